// MolecularGNNEncoder_74990128988469
// MI455X (gfx1250) — compile-verified
//
#include <hip/hip_runtime.h>
#include <hip/hip_bf16.h>

// ---------------------------------------------------------------------------
// MolecularGNNEncoder for MI455X (gfx1250, wave32)
//
// Bandwidth-bound workload (~12 GFLOP vs ~2 GB of edge gather/scatter traffic
// -> ~90us floor at 23.3 TB/s). Edge embeddings are recomputed from the 24 B
// raw attrs instead of materializing [E,64] (saves ~600 MB of traffic); the
// layer-0 edge projection composes with GINE's lin() into one 6->11 affine.
// Dense node/graph MLPs run on V_WMMA_F32_16X16X4_F32 with compile-time
// dimensions so all K-loops fully unroll into straight-line WMMA streams.
// ---------------------------------------------------------------------------

typedef __attribute__((ext_vector_type(2))) float v2f;
typedef __attribute__((ext_vector_type(8))) float v8f;

#if defined(__gfx1250__) && __has_builtin(__builtin_amdgcn_wmma_f32_16x16x4_f32)
#define USE_WMMA_F32 1
#else
#define USE_WMMA_F32 0
#endif

// One 16x16x4 K-step of D += A*B for a 16x16 f32 tile.
// Fragment layouts per CDNA5 ISA 7.12.2:
//   A 16x4 : lanes 0-15 hold {K=k0,k0+1}, lanes 16-31 hold {K=k0+2,k0+3}, M = lane%16
//   B 4x16 : lanes 0-15 hold rows {k0,k0+1}, lanes 16-31 rows {k0+2,k0+3}, N = n0+lane%16
//   C/D    : VGPR r -> M = r + (lane<16 ? 0 : 8), N = n0 + lane%16
// With constant k0/in_dim (unrolled callers) the guards fold at compile time.
__device__ __forceinline__ void wmma_acc_tile(v8f& c,
                                              const float* lds_t, int t_stride,
                                              const float* __restrict__ W, int w_ld,
                                              int n0, int k0, int in_dim, int lane) {
    const int half = lane >> 4;       // 0: lanes 0-15, 1: lanes 16-31
    const int l16  = lane & 15;
    const int col  = n0 + l16;
#if USE_WMMA_F32
    const int kk = k0 + half * 2;
    v2f a, b;
    a.x = lds_t[l16 * t_stride + kk];
    a.y = lds_t[l16 * t_stride + kk + 1];
    b.x = (kk     < in_dim) ? W[(size_t)kk       * w_ld + col] : 0.0f;
    b.y = (kk + 1 < in_dim) ? W[(size_t)(kk + 1) * w_ld + col] : 0.0f;
    c = __builtin_amdgcn_wmma_f32_16x16x4_f32(false, a, false, b,
                                              (short)0, c, false, false);
#else
    float w0 = (k0     < in_dim) ? W[(size_t)(k0    ) * w_ld + col] : 0.0f;
    float w1 = (k0 + 1 < in_dim) ? W[(size_t)(k0 + 1) * w_ld + col] : 0.0f;
    float w2 = (k0 + 2 < in_dim) ? W[(size_t)(k0 + 2) * w_ld + col] : 0.0f;
    float w3 = (k0 + 3 < in_dim) ? W[(size_t)(k0 + 3) * w_ld + col] : 0.0f;
#pragma unroll
    for (int r = 0; r < 8; ++r) {
        const float* tr = lds_t + (r + half * 8) * t_stride + k0;
        c[r] += tr[0] * w0 + tr[1] * w1 + tr[2] * w2 + tr[3] * w3;
    }
#endif
}

// Fused 2-layer row MLP: Out = act2(bn2(relu(bn1((X+X2)@W1+B1))@W2+B2))
// One block = 16 rows, 128 threads = 4 waves. MID = 128 in all uses.
template <int IN_DIM, int OUT_DIM, bool BN1, bool BN2, bool RELU2, bool HAS_X2>
__global__ __launch_bounds__(128) void fused_mlp16_kernel(
    const float* __restrict__ X, const float* __restrict__ X2,
    const float* __restrict__ W1, const float* __restrict__ B1,
    const float* __restrict__ bn1g, const float* __restrict__ bn1b,
    const float* __restrict__ bn1m, const float* __restrict__ bn1v,
    const float* __restrict__ W2, const float* __restrict__ B2,
    const float* __restrict__ bn2g, const float* __restrict__ bn2b,
    const float* __restrict__ bn2m, const float* __restrict__ bn2v,
    float* __restrict__ Out, int Mrows) {
    constexpr int MID      = 128;
    constexpr int IN_PAD   = (IN_DIM + 3) & ~3;
    constexpr int T_STRIDE = IN_PAD + 1;   // +1: kill LDS bank conflicts on A reads
    constexpr int M_STRIDE = MID + 1;

    __shared__ float lds_t[16 * T_STRIDE];
    __shared__ float lds_m[16 * M_STRIDE];

    const int tid  = threadIdx.x;
    const int lane = tid & 31;
    const int wave = tid >> 5;
    const int m0   = blockIdx.x * 16;

    // Stage t = X (+ X2) into LDS, zero-padding K to a multiple of 4.
    for (int idx = tid; idx < 16 * T_STRIDE; idx += 128) {
        int row = idx / T_STRIDE, col = idx % T_STRIDE;
        int gr  = m0 + row;
        float v = 0.0f;
        if (col < IN_DIM && gr < Mrows) {
            v = X[(size_t)gr * IN_DIM + col];
            if constexpr (HAS_X2) v += X2[(size_t)gr * IN_DIM + col];
        }
        lds_t[idx] = v;
    }
    __syncthreads();

    const int half = lane >> 4;
    const int l16  = lane & 15;

    // Layer 1: MID/16 = 8 tiles -> 2 per wave; K fully unrolled.
#pragma unroll
    for (int nt = wave; nt < (MID >> 4); nt += 4) {
        const int n0 = nt * 16;
        v8f c = {0.f, 0.f, 0.f, 0.f, 0.f, 0.f, 0.f, 0.f};
#pragma unroll
        for (int k0 = 0; k0 < IN_PAD; k0 += 4)
            wmma_acc_tile(c, lds_t, T_STRIDE, W1, MID, n0, k0, IN_DIM, lane);
        const int col  = n0 + l16;
        const float bias = B1[col];
        float scale = 1.0f, shift = 0.0f;
        if constexpr (BN1) {
            float inv = rsqrtf(bn1v[col] + 1e-5f);
            scale = bn1g[col] * inv;
            shift = bn1b[col] - bn1m[col] * scale;
        }
#pragma unroll
        for (int r = 0; r < 8; ++r) {
            float v = (c[r] + bias) * scale + shift;
            v = fmaxf(v, 0.0f);                       // relu1 always on
            lds_m[(r + half * 8) * M_STRIDE + col] = v;
        }
    }
    __syncthreads();

    // Layer 2: OUT_DIM/16 tiles -> 1 or 2 per wave; K = 128 fully unrolled.
#pragma unroll
    for (int nt = wave; nt < (OUT_DIM >> 4); nt += 4) {
        const int n0 = nt * 16;
        v8f c = {0.f, 0.f, 0.f, 0.f, 0.f, 0.f, 0.f, 0.f};
#pragma unroll
        for (int k0 = 0; k0 < MID; k0 += 4)
            wmma_acc_tile(c, lds_m, M_STRIDE, W2, OUT_DIM, n0, k0, MID, lane);
        const int col  = n0 + l16;
        const float bias = B2[col];
        float scale = 1.0f, shift = 0.0f;
        if constexpr (BN2) {
            float inv = rsqrtf(bn2v[col] + 1e-5f);
            scale = bn2g[col] * inv;
            shift = bn2b[col] - bn2m[col] * scale;
        }
#pragma unroll
        for (int r = 0; r < 8; ++r) {
            int gr  = m0 + r + half * 8;
            float v = (c[r] + bias) * scale + shift;
            if constexpr (RELU2) v = fmaxf(v, 0.0f);
            if (gr < Mrows) Out[(size_t)gr * OUT_DIM + col] = v;
        }
    }
}

// Compose layer-0's edge path: Wf = edge_w @ lin_w, bf = edge_b @ lin_w + lin_b.
__global__ void prep_fused_edge_kernel(const float* __restrict__ ew,  // [6,64]
                                       const float* __restrict__ eb,  // [64]
                                       const float* __restrict__ lw,  // [64,11]
                                       const float* __restrict__ lb,  // [11]
                                       float* __restrict__ Wf,        // [6,11]
                                       float* __restrict__ bf) {      // [11]
    int t = threadIdx.x;
    if (t < 66) {
        int k = t / 11, j = t % 11;
        float s = 0.0f;
        for (int c = 0; c < 64; ++c) s += ew[k * 64 + c] * lw[c * 11 + j];
        Wf[k * 11 + j] = s;
    } else if (t < 77) {
        int j = t - 66;
        float s = lb[j];
        for (int c = 0; c < 64; ++c) s += eb[c] * lw[c * 11 + j];
        bf[j] = s;
    }
}

// GINE message + aggregate: one wave per edge; edge embedding recomputed from
// the 6 raw attrs (saves materializing a 307 MB [E,64] tensor).
template <int DIM>
__global__ __launch_bounds__(256) void gine_msg_kernel(
    const float* __restrict__ h,    // [N, DIM]
    const int*   __restrict__ ei,   // [2, E]: src = ei[0..E), dst = ei[E..2E)
    const float* __restrict__ ea,   // [E, 6]
    const float* __restrict__ W,    // [6, DIM]
    const float* __restrict__ b,    // [DIM]
    float* __restrict__ agg,        // [N, DIM]
    int E) {
    const int wid  = (int)((blockIdx.x * (unsigned)blockDim.x + threadIdx.x) >> 5);
    const int lane = threadIdx.x & 31;
    if (wid >= E) return;

    if (wid + 4096 < E)   // pull upcoming attr rows toward L2 (global_prefetch_b8)
        __builtin_prefetch(&ea[(size_t)(wid + 4096) * 6], 0, 1);

    const int src = ei[wid];
    const int dst = ei[E + wid];
    const float* a = &ea[(size_t)wid * 6];
    const float a0 = a[0], a1 = a[1], a2 = a[2], a3 = a[3], a4 = a[4], a5 = a[5];
    const float* hs = &h[(size_t)src * DIM];
    float* ag = &agg[(size_t)dst * DIM];

#pragma unroll
    for (int c = lane; c < DIM; c += 32) {
        float e = b[c] + a0 * W[c] + a1 * W[DIM + c] + a2 * W[2 * DIM + c] +
                  a3 * W[3 * DIM + c] + a4 * W[4 * DIM + c] + a5 * W[5 * DIM + c];
        float v = fmaxf(hs[c] + e, 0.0f);
        atomicAdd(&ag[c], v);
    }
}

// Graph pooling scatter: one wave per node.
__global__ __launch_bounds__(256) void pool_scatter_kernel(
    const float* __restrict__ h, const int* __restrict__ batch,
    float* __restrict__ sums, float* __restrict__ cnt, int N) {
    const int wid  = (int)((blockIdx.x * (unsigned)blockDim.x + threadIdx.x) >> 5);
    const int lane = threadIdx.x & 31;
    if (wid >= N) return;
    const int g = batch[wid];
#pragma unroll
    for (int c = lane; c < 64; c += 32)
        atomicAdd(&sums[(size_t)g * 64 + c], h[(size_t)wid * 64 + c]);
    if (lane == 0) atomicAdd(&cnt[g], 1.0f);
}

// pooled[g] = [ mean(h) | sum(h) ]  -> [G,128]
__global__ __launch_bounds__(256) void build_pooled_kernel(
    const float* __restrict__ sums, const float* __restrict__ cnt,
    float* __restrict__ pooled, int G) {
    int i = blockIdx.x * blockDim.x + threadIdx.x;
    if (i >= G * 128) return;
    int g = i >> 7, c = i & 127;
    float s = (c < 64) ? sums[g * 64 + c] / fmaxf(cnt[g], 1.0f)
                       : sums[g * 64 + (c - 64)];
    pooled[i] = s;
}

// ---------------------------------------------------------------------------
// Host launch. Input flattening assumed in setup_inputs() insertion order:
//  0:x 1:edge_index 2:edge_attr 3:batch 4:n_graphs 5:edge_w 6:edge_b
//  7..20:  layer0 {w1,b1,w2,b2,bn1_g,bn1_b,bn1_m,bn1_v,bn_g,bn_b,bn_m,bn_v,lin_w,lin_b}
//  21..32: layer1 (12)   33..44: layer2 (12)
//  45:proj_w1 46:proj_b1 47:proj_w2 48:proj_b2
// ---------------------------------------------------------------------------
extern "C" void kernel_launch(void* const* d_in, const int* in_sizes, int n_in,
                              void* d_out, int out_size, void* d_ws, size_t ws_size,
                              hipStream_t stream) {
    (void)n_in; (void)ws_size;
    auto P = [&](int i) { return (const float*)d_in[i]; };

    const float* x     = P(0);
    const int*   ei    = (const int*)d_in[1];
    const float* ea    = P(2);
    const int*   batch = (const int*)d_in[3];
    const float* ew    = P(5);
    const float* ebias = P(6);

    const int N = in_sizes[0] / 11;
    const int E = in_sizes[2] / 6;
    const int G = out_size / 128;

    // Workspace layout (~80 MB)
    char* ws = (char*)d_ws;
    const size_t hbytes = (size_t)N * 64 * sizeof(float);
    float* agg  = (float*)(ws);
    float* hA   = (float*)(ws + hbytes);
    float* hB   = (float*)(ws + 2 * hbytes);
    float* Wf   = (float*)(ws + 3 * hbytes);          // 66 floats
    float* bf   = Wf + 66;                            // 11 floats
    float* sums = (float*)(ws + 3 * hbytes + 512);    // G*64
    float* cnt  = (float*)((char*)sums + (size_t)G * 64 * sizeof(float));
    float* pooled = (float*)(((uintptr_t)cnt + (size_t)G * sizeof(float) + 255) & ~(uintptr_t)255);

    const int msgBlocks  = (int)(((size_t)E * 32 + 255) / 256);
    const int poolBlocks = (int)(((size_t)N * 32 + 255) / 256);

    // 0) Compose layer-0 edge affine (6 -> 11).
    prep_fused_edge_kernel<<<1, 128, 0, stream>>>(ew, ebias, P(19), P(20), Wf, bf);

    // Layer 0
    hipMemsetAsync(agg, 0, hbytes, stream);
    gine_msg_kernel<11><<<msgBlocks, 256, 0, stream>>>(x, ei, ea, Wf, bf, agg, E);
    fused_mlp16_kernel<11, 64, true, true, true, true><<<N / 16, 128, 0, stream>>>(
        x, agg, P(7), P(8), P(11), P(12), P(13), P(14),
        P(9), P(10), P(15), P(16), P(17), P(18), hA, N);

    // Layer 1
    hipMemsetAsync(agg, 0, hbytes, stream);
    gine_msg_kernel<64><<<msgBlocks, 256, 0, stream>>>(hA, ei, ea, ew, ebias, agg, E);
    fused_mlp16_kernel<64, 64, true, true, true, true><<<N / 16, 128, 0, stream>>>(
        hA, agg, P(21), P(22), P(25), P(26), P(27), P(28),
        P(23), P(24), P(29), P(30), P(31), P(32), hB, N);

    // Layer 2
    hipMemsetAsync(agg, 0, hbytes, stream);
    gine_msg_kernel<64><<<msgBlocks, 256, 0, stream>>>(hB, ei, ea, ew, ebias, agg, E);
    fused_mlp16_kernel<64, 64, true, true, true, true><<<N / 16, 128, 0, stream>>>(
        hB, agg, P(33), P(34), P(37), P(38), P(39), P(40),
        P(35), P(36), P(41), P(42), P(43), P(44), hA, N);

    // Pooling
    hipMemsetAsync(sums, 0, (size_t)G * 64 * sizeof(float), stream);
    hipMemsetAsync(cnt, 0, (size_t)G * sizeof(float), stream);
    pool_scatter_kernel<<<poolBlocks, 256, 0, stream>>>(hA, batch, sums, cnt, N);
    build_pooled_kernel<<<(G * 128 + 255) / 256, 256, 0, stream>>>(sums, cnt, pooled, G);

    // Projection head (no BN; relu after first layer only)
    fused_mlp16_kernel<128, 128, false, false, false, false><<<G / 16, 128, 0, stream>>>(
        pooled, nullptr, P(45), P(46), nullptr, nullptr, nullptr, nullptr,
        P(47), P(48), nullptr, nullptr, nullptr, nullptr,
        (float*)d_out, G);
}